// CondConv1x1_49151605736014
// MI455X (gfx1250) — compile-verified
//
#include <hip/hip_runtime.h>
#include <math.h>

// CondConv1x1 for MI455X (gfx1250):
//   weight = tanh(tanh(MLP(cond)))  [B,64,64]
//   z      = weight @ x per batch   (WMMA f32 16x16x4; X staged in LDS in a
//            K-pair-interleaved layout so each B fragment is one ds_load_b64;
//            2 independent accumulators keep the XDL pipe free of RAW chains)
//   logdet = slogdet(weight)*H*W + log_det   (LU w/ partial pivoting)

typedef __attribute__((ext_vector_type(2))) float v2f;
typedef __attribute__((ext_vector_type(8))) float v8f;

#define CCH 64          // channels
#define NB  32          // batch
#define HW  16384       // H*W
#define NCHUNK 128      // N columns staged in LDS per workgroup
// LDS layout: element (k, n) lives at float offset (k>>1)*PRS + 2*n + (k&1).
// PRS = 2*NCHUNK + 32: the +32 pad makes consecutive k-pair rows shift by 32
// dword-banks, so the two half-waves (K and K+2) read disjoint bank ranges.
#define PRS (2 * NCHUNK + 32)

// ---------------------------------------------------------------------------
// Kernel 1: cond_net  (Linear+ReLU, Linear+ReLU, Linear+tanh, then tanh again)
// one workgroup per sample; tiny GEMMs, weights live in L2.
// ---------------------------------------------------------------------------
__global__ __launch_bounds__(256) void cond_net_kernel(
    const float* __restrict__ cond,
    const float* __restrict__ W1, const float* __restrict__ b1,
    const float* __restrict__ W2, const float* __restrict__ b2,
    const float* __restrict__ W3, const float* __restrict__ b3,
    float* __restrict__ wgt)
{
    __shared__ float cond_s[128];
    __shared__ float h1[512];
    __shared__ float h2[512];
    const int b = blockIdx.x;
    const int t = threadIdx.x;

    if (t < 128) cond_s[t] = cond[b * 128 + t];
    __syncthreads();

    for (int j = t; j < 512; j += 256) {
        float s = b1[j];
        const float* w = W1 + (size_t)j * 128;
        #pragma unroll 8
        for (int k = 0; k < 128; ++k) s = fmaf(cond_s[k], w[k], s);
        h1[j] = fmaxf(s, 0.f);
    }
    __syncthreads();

    for (int j = t; j < 512; j += 256) {
        float s = b2[j];
        const float* w = W2 + (size_t)j * 512;
        #pragma unroll 8
        for (int k = 0; k < 512; ++k) s = fmaf(h1[k], w[k], s);
        h2[j] = fmaxf(s, 0.f);
    }
    __syncthreads();

    for (int j = t; j < 4096; j += 256) {
        float s = b3[j];
        const float* w = W3 + (size_t)j * 512;
        #pragma unroll 8
        for (int k = 0; k < 512; ++k) s = fmaf(h2[k], w[k], s);
        wgt[(size_t)b * 4096 + j] = tanhf(tanhf(s));   // get_weight applies tanh twice
    }
}

// ---------------------------------------------------------------------------
// Kernel 2: z[b] = weight[b] (64x64) x X[b] (64x16384), v_wmma_f32_16x16x4_f32
// grid = (HW/NCHUNK, NB), 256 threads = 8 waves.
// ---------------------------------------------------------------------------
__global__ __launch_bounds__(256) void mix_kernel(
    const float* __restrict__ x,
    const float* __restrict__ wgt,
    float* __restrict__ z)
{
    __shared__ float xs[(CCH / 2) * PRS];     // 32 k-pair rows * 288 floats = 36 KB

    const int b    = blockIdx.y;
    const int n0   = blockIdx.x * NCHUNK;
    const int t    = threadIdx.x;
    const int lane = t & 31;
    const int wave = t >> 5;

    // --- cooperative stage: interleave rows (2p, 2p+1) column-wise ---------
    // thread task (p, j): read float2 of row 2p and row 2p+1 at columns {2j,2j+1},
    // write {x[2p][2j], x[2p+1][2j], x[2p][2j+1], x[2p+1][2j+1]} as one b128.
    const float* xb = x + (size_t)b * CCH * HW;
    for (int i = t; i < (CCH / 2) * (NCHUNK / 2); i += 256) {   // 2048 tasks, 8 iters
        const int p = i / (NCHUNK / 2);
        const int j = i % (NCHUNK / 2);
        const float2 g0 = *(const float2*)(xb + (size_t)(2 * p)     * HW + n0 + 2 * j);
        const float2 g1 = *(const float2*)(xb + (size_t)(2 * p + 1) * HW + n0 + 2 * j);
        float4 v;
        v.x = g0.x; v.y = g1.x; v.z = g0.y; v.w = g1.y;
        *(float4*)(&xs[p * PRS + 4 * j]) = v;
    }

    // --- A fragments (weight 16x64 slab for this wave's M-tile), resident ---
    // 16x4 f32 A layout: lanes 0-15 M=0..15; VGPR0 = K{0 | 2}, VGPR1 = K{1 | 3}
    const int mtile = wave >> 1;               // waves {0,1}->m0, {2,3}->m1, ...
    const int mrow  = mtile * 16 + (lane & 15);
    const int khalf = (lane >> 4) * 2;         // 0 for lanes 0-15, 2 for 16-31
    v2f a[16];
    const float* wb = wgt + (size_t)b * 4096 + (size_t)mrow * 64;
    #pragma unroll
    for (int kb = 0; kb < 16; ++kb) {
        const float2 av = *(const float2*)(wb + 4 * kb + khalf);  // one b64 per block
        a[kb].x = av.x; a[kb].y = av.y;
    }
    __syncthreads();

    // --- WMMA over 4 N-tiles per wave with independent accumulators --------
    // B 4x16 fragment = {K=r, K=r+1} at column n -> contiguous b64 in LDS:
    //   float offset = (2*kb + lane>>4)*PRS + 2*(nn + ncol)
    float* zb = z + (size_t)b * CCH * HW + (size_t)mtile * 16 * HW + n0;
    const int ncol  = lane & 15;
    const int nbase = (wave & 1) * 4;
    const int moff  = (lane >> 4) * 8;
    const int prow0 = lane >> 4;               // k-pair sub-row for this half-wave
    #pragma unroll
    for (int pair = 0; pair < 2; ++pair) {
        const int nn0 = (nbase + pair * 2) * 16;
        const int nn1 = nn0 + 16;
        const float* ls0 = &xs[prow0 * PRS + 2 * (nn0 + ncol)];
        const float* ls1 = &xs[prow0 * PRS + 2 * (nn1 + ncol)];
        v8f acc0 = {};
        v8f acc1 = {};
        #pragma unroll
        for (int kb = 0; kb < 16; ++kb) {
            const v2f bf0 = *(const v2f*)(ls0 + 2 * kb * PRS);   // ds_load_b64
            const v2f bf1 = *(const v2f*)(ls1 + 2 * kb * PRS);   // ds_load_b64
            acc0 = __builtin_amdgcn_wmma_f32_16x16x4_f32(
                false, a[kb], false, bf0, (short)0, acc0, false, false);
            acc1 = __builtin_amdgcn_wmma_f32_16x16x4_f32(
                false, a[kb], false, bf1, (short)0, acc1, false, false);
        }
        // C/D 16x16 f32: VGPR v -> M = v (lanes 0-15) or v+8 (lanes 16-31), N = lane&15
        #pragma unroll
        for (int v = 0; v < 8; ++v) {
            zb[(size_t)(moff + v) * HW + nn0 + ncol] = acc0[v];
            zb[(size_t)(moff + v) * HW + nn1 + ncol] = acc1[v];
        }
    }
}

// ---------------------------------------------------------------------------
// Kernel 3: logabsdet(weight[b]) via LU with partial pivoting; 64 threads/block.
// new_log_det = log_det + logabsdet * (H*W)
// ---------------------------------------------------------------------------
__global__ __launch_bounds__(64) void slogdet_kernel(
    const float* __restrict__ wgt,
    const float* __restrict__ log_det_in,
    float* __restrict__ log_det_out)
{
    __shared__ float a[64][65];     // +1 pad against bank conflicts
    __shared__ float acc_s;
    const int b = blockIdx.x;
    const int t = threadIdx.x;

    for (int r = 0; r < 64; ++r) a[r][t] = wgt[(size_t)b * 4096 + r * 64 + t];
    if (t == 0) acc_s = 0.f;
    __syncthreads();

    for (int p = 0; p < 64; ++p) {
        if (t == 0) {
            int pr = p; float pv = fabsf(a[p][p]);
            for (int r = p + 1; r < 64; ++r) {
                const float v = fabsf(a[r][p]);
                if (v > pv) { pv = v; pr = r; }
            }
            if (pr != p)
                for (int j = p; j < 64; ++j) {
                    const float tmp = a[p][j]; a[p][j] = a[pr][j]; a[pr][j] = tmp;
                }
            acc_s += logf(fabsf(a[p][p]));
        }
        __syncthreads();
        const int r = p + 1 + t;
        if (r < 64) {
            const float f = a[r][p] / a[p][p];
            for (int j = p + 1; j < 64; ++j)
                a[r][j] = fmaf(-f, a[p][j], a[r][j]);
        }
        __syncthreads();
    }
    if (t == 0) log_det_out[b] = log_det_in[b] + acc_s * (float)HW;
}

// ---------------------------------------------------------------------------
extern "C" void kernel_launch(void* const* d_in, const int* in_sizes, int n_in,
                              void* d_out, int out_size, void* d_ws, size_t ws_size,
                              hipStream_t stream) {
    const float* x    = (const float*)d_in[0];
    const float* cond = (const float*)d_in[1];
    const float* ld   = (const float*)d_in[2];
    const float* W1   = (const float*)d_in[3];
    const float* b1   = (const float*)d_in[4];
    const float* W2   = (const float*)d_in[5];
    const float* b2   = (const float*)d_in[6];
    const float* W3   = (const float*)d_in[7];
    const float* b3   = (const float*)d_in[8];

    float* z      = (float*)d_out;
    float* ld_out = z + (size_t)NB * CCH * HW;     // outputs concatenated flat
    float* wgt    = (float*)d_ws;                  // [32,64,64] = 512 KB scratch

    cond_net_kernel<<<NB, 256, 0, stream>>>(cond, W1, b1, W2, b2, W3, b3, wgt);

    dim3 grid(HW / NCHUNK, NB);                    // 128 x 32 workgroups
    mix_kernel<<<grid, 256, 0, stream>>>(x, wgt, z);

    slogdet_kernel<<<NB, 64, 0, stream>>>(wgt, ld, ld_out);
}